// SelfMixing_31791347925868
// MI455X (gfx1250) — compile-verified
//
#include <hip/hip_runtime.h>
#include <math.h>

// ================= static geometry (matches reference exactly) =================
// METADATA = [64,48,32], MAX_L=3
// OFF_IN  = [0, 64, 208, 368]   DIM_IN  = 368
// OFF_OUT = [0, 64, 208, 448, 672] (MD_OV = [64,48,48,32]) DIM_OUT = 672
// PATHS (l1-major order) with mix-coeff bases and dense-CG offsets:
#define NPATH 18
__device__ const int P_L1[NPATH]  = {0,0,0,1,1,1,1,1,1,1,2,2,2,2,2,2,2,2};
__device__ const int P_L2[NPATH]  = {0,1,2,0,1,1,1,2,2,2,0,1,1,1,2,2,2,2};
__device__ const int P_L3[NPATH]  = {0,1,2,1,0,1,2,1,2,3,2,1,2,3,0,1,2,3};
__device__ const int P_OFF[NPATH] = {0,1,10,35,44,53,80,125,170,245,350,375,420,495,600,625,700,825};
#define CG_TOTAL 1000

// ================= setup kernel: build real CG tables on device ================
__device__ __forceinline__ double dfact(int n) {
  const double f[9] = {1.,1.,2.,6.,24.,120.,720.,5040.,40320.};
  return f[n];
}

// Racah formula, identical to the reference's _cg_complex (valid l ranges only)
__device__ double cg_cplx(int l1,int m1,int l2,int m2,int l3,int m3) {
  if (m1 + m2 != m3) return 0.0;
  double pre = sqrt((2.0*l3+1.0)*dfact(l1+l2-l3)*dfact(l1-l2+l3)*dfact(-l1+l2+l3)/dfact(l1+l2+l3+1));
  pre *= sqrt(dfact(l1+m1)*dfact(l1-m1)*dfact(l2+m2)*dfact(l2-m2)*dfact(l3+m3)*dfact(l3-m3));
  int kmin = 0;
  if (l2-l3-m1 > kmin) kmin = l2-l3-m1;
  if (l1-l3+m2 > kmin) kmin = l1-l3+m2;
  int kmax = l1+l2-l3;
  if (l1-m1 < kmax) kmax = l1-m1;
  if (l2+m2 < kmax) kmax = l2+m2;
  double s = 0.0;
  for (int k = kmin; k <= kmax; ++k) {
    double d = dfact(k)*dfact(l1+l2-l3-k)*dfact(l1-m1-k)*dfact(l2+m2-k)*dfact(l3-l2+m1+k)*dfact(l3-l1-m2+k);
    s += (k & 1) ? (-1.0/d) : (1.0/d);
  }
  return pre * s;
}

struct Cplx { double re, im; };
// One entry of the complex->real change-of-basis U(l), identical to reference _U
__device__ __forceinline__ Cplx uent(int l, int mr, int mc) {
  const double is2 = 0.70710678118654752440;
  Cplx u; u.re = 0.0; u.im = 0.0;
  if (mr > 0)      { if (mc ==  mr) u.re = (mr & 1) ? -is2 : is2; else if (mc == -mr) u.re = is2; }
  else if (mr == 0){ if (mc == 0) u.re = 1.0; }
  else             { if (mc ==  mr) u.im = is2; else if (mc == -mr) u.im = ((-mr) & 1) ? is2 : -is2; }
  return u;
}
__device__ __forceinline__ Cplx cmul(Cplx a, Cplx b) {
  Cplx r; r.re = a.re*b.re - a.im*b.im; r.im = a.re*b.im + a.im*b.re; return r;
}

__global__ void selfmix_cg_build(float* __restrict__ cg_out) {
  const int p  = blockIdx.x;
  const int l1 = P_L1[p], l2 = P_L2[p], l3 = P_L3[p];
  const int N2 = 2*l2+1, N3 = 2*l3+1;
  const int cnt = (2*l1+1)*N2*N3;
  const int odd = (l1 + l2 + l3) & 1;   // odd parity => real CG lives in imag part
  for (int e = threadIdx.x; e < cnt; e += blockDim.x) {
    const int cI = e % N3; const int ab = e / N3;
    const int b = ab % N2; const int a = ab / N2;
    const int mr1 = a - l1, mr2 = b - l2, mr3 = cI - l3;
    const int n1 = (mr1 == 0) ? 1 : 2, n2 = (mr2 == 0) ? 1 : 2, n3 = (mr3 == 0) ? 1 : 2;
    int mc1s[2] = { mr1, -mr1 }, mc2s[2] = { mr2, -mr2 }, mc3s[2] = { mr3, -mr3 };
    double re = 0.0, im = 0.0;
    for (int i = 0; i < n1; ++i)
      for (int j = 0; j < n2; ++j)
        for (int k = 0; k < n3; ++k) {
          Cplx u1 = uent(l1, mr1, mc1s[i]);
          Cplx u2 = uent(l2, mr2, mc2s[j]);
          Cplx u3 = uent(l3, mr3, mc3s[k]); u3.im = -u3.im;  // conj(U(l3))
          double cc = cg_cplx(l1, mc1s[i], l2, mc2s[j], l3, mc3s[k]);
          Cplx t = cmul(cmul(u1, u2), u3);
          re += t.re * cc; im += t.im * cc;
        }
    cg_out[P_OFF[p] + e] = (float)(odd ? im : re);
  }
}

// ================= main kernel ================================================
// Exact necessary condition for a real-CG entry to be nonzero:
//  |m3| in {|m1|+|m2|, ||m1|-|m2||}  AND  (#negative m's) parity == (l1+l2+l3) parity.
// Constant-folds after full unroll -> dead FMAs eliminated (211 of 1000 survive).
__device__ constexpr bool nz_cg(int l1,int l2,int l3,int a,int b,int c) {
  int m1 = a-l1, m2 = b-l2, m3 = c-l3;
  int am1 = m1<0?-m1:m1, am2 = m2<0?-m2:m2, am3 = m3<0?-m3:m3;
  int s = am1+am2, d = am1>am2 ? am1-am2 : am2-am1;
  if (am3 != s && am3 != d) return false;
  int neg = (m1<0) + (m2<0) + (m3<0);
  return (neg & 1) == ((l1+l2+l3) & 1);
}

template<int L1,int L2,int L3>
__device__ __forceinline__ void path_tp(const float* cg, const float* x1, const float* x2,
                                        float mixv, float* out) {
  constexpr int N1 = 2*L1+1, N2 = 2*L2+1, N3 = 2*L3+1;
  float acc[N3];
  #pragma unroll
  for (int m = 0; m < N3; ++m) acc[m] = 0.0f;
  #pragma unroll
  for (int a = 0; a < N1; ++a) {
    #pragma unroll
    for (int b = 0; b < N2; ++b) {
      const float t = x1[a] * x2[b];   // DCE'd when no surviving m
      #pragma unroll
      for (int m = 0; m < N3; ++m) {
        if (nz_cg(L1, L2, L3, a, b, m))
          acc[m] = fmaf(cg[(a*N2+b)*N3 + m], t, acc[m]);
      }
    }
  }
  #pragma unroll
  for (int m = 0; m < N3; ++m) out[m] = fmaf(acc[m], mixv, out[m]);
}

typedef unsigned int v4u __attribute__((ext_vector_type(4)));
typedef int          v8i __attribute__((ext_vector_type(8)));
typedef int          v4i __attribute__((ext_vector_type(4)));

__global__ __launch_bounds__(256) void selfmix_main(
    const float* __restrict__ x,     // [nrows,368]
    const float* __restrict__ keep,  // [144]
    const float* __restrict__ mix,   // [688]
    const float* __restrict__ cg_g,  // [1000] dense CG (d_ws)
    float* __restrict__ out,         // [nrows,672]
    int nrows)
{
  __shared__ float cg[1024];

  // ---- Stage CG table via CDNA5 Tensor Data Mover (wave 0 only; TDM done
  // signal tracked by TENSORcnt). 1D tile: 1000 x f32, global -> LDS offset 0.
  // This toolchain exposes the 6-arg builtin:
  //   (uint32x4 g0, int32x8 g1, int32x4 g2, int32x4 g3, int32x8 g4, i32 cpol)
#if __has_builtin(__builtin_amdgcn_tensor_load_to_lds)
  if (threadIdx.x < 32u) {
    unsigned long long ga = (unsigned long long)(size_t)cg_g;
    v4u g0;
    g0.x = 1u;                                   // count=1, user mode
    g0.y = 0u;                                   // lds_addr = 0 (cg[] is sole LDS alloc)
    g0.z = (unsigned)(ga & 0xFFFFFFFFull);       // global_addr[31:0]
    g0.w = (unsigned)((ga >> 32) & 0x1FFFFFFull) | (2u << 30);  // addr[56:32] | type=2
    v8i g1;
    g1[0] = 0x00020000;      // workgroup_mask=0, data_size=2 (4B)
    g1[1] = 1000 << 16;      // tensor_dim0 = 1000
    g1[2] = 1 << 16;         // tensor_dim1 = 1
    g1[3] = 1000 << 16;      // tile_dim0 = 1000
    g1[4] = 1;               // tile_dim1 = 1, tile_dim2 = 0
    g1[5] = 1000;            // tensor_dim0_stride lo
    g1[6] = 1000 << 16;      // stride hi=0 | tensor_dim1_stride lo = 1000
    g1[7] = 0;
    v4i g2 = {0,0,0,0}, g3 = {0,0,0,0};          // groups 2/3 unused (<=2D tensor)
    v8i g4 = {0,0,0,0,0,0,0,0};                  // extra group operand: zero-filled
    __builtin_amdgcn_tensor_load_to_lds(g0, g1, g2, g3, g4, 0);
  #if __has_builtin(__builtin_amdgcn_s_wait_tensorcnt)
    __builtin_amdgcn_s_wait_tensorcnt(0);
  #else
    asm volatile("s_wait_tensorcnt 0x0" ::: "memory");
  #endif
  }
#endif
  // Authoritative cooperative copy (runs after the TDM wait; same bytes, so
  // correctness never depends on D# packing details).
  for (int i = threadIdx.x; i < CG_TOTAL; i += 256) cg[i] = cg_g[i];
  __syncthreads();

  const int  t   = threadIdx.x;
  const int  c   = t & 63;                      // channel (lane-contiguous => coalesced)
  const int  row = (int)blockIdx.x * 4 + (t >> 6);
  const bool active = row < nrows;
  const bool has1   = active && (c < 48);
  const bool has2   = active && (c < 32);

  const float* xr = x + (size_t)row * 368;
  float x0 = 0.0f, x1[3] = {0,0,0}, x2[5] = {0,0,0,0,0};
  if (active) x0 = xr[c];
  if (has1) {
    #pragma unroll
    for (int m = 0; m < 3; ++m) x1[m] = xr[64 + c*3 + m];
  }
  if (has2) {
    #pragma unroll
    for (int m = 0; m < 5; ++m) x2[m] = xr[208 + c*5 + m];
  }

  float xs[1] = { x0 };
  float o0[1] = { 0.0f };
  float o1[3] = {0,0,0}, o2[5] = {0,0,0,0,0}, o3[7] = {0,0,0,0,0,0,0};

  if (active) {
    path_tp<0,0,0>(cg + 0,   xs, xs, 0.5f*mix[  0 + c], o0);
    o0[0] = fmaf(keep[c], x0, o0[0]);                      // keep, l=0
  }
  if (has1) {
    path_tp<0,1,1>(cg + 1,   xs, x1, 0.5f*mix[ 64 + c], o1);
    path_tp<1,0,1>(cg + 35,  x1, xs, 0.5f*mix[144 + c], o1);
    path_tp<1,1,0>(cg + 44,  x1, x1, 0.5f*mix[192 + c], o0);
    path_tp<1,1,1>(cg + 53,  x1, x1, 0.5f*mix[240 + c], o1);
    path_tp<1,1,2>(cg + 80,  x1, x1, 0.5f*mix[288 + c], o2);
    const float kc = keep[64 + c];                         // keep, l=1
    #pragma unroll
    for (int m = 0; m < 3; ++m) o1[m] = fmaf(kc, x1[m], o1[m]);
  }
  if (has2) {
    path_tp<0,2,2>(cg + 10,  xs, x2, 0.5f*mix[112 + c], o2);
    path_tp<1,2,1>(cg + 125, x1, x2, 0.5f*mix[336 + c], o1);
    path_tp<1,2,2>(cg + 170, x1, x2, 0.5f*mix[368 + c], o2);
    path_tp<1,2,3>(cg + 245, x1, x2, 0.5f*mix[400 + c], o3);
    path_tp<2,0,2>(cg + 350, x2, xs, 0.5f*mix[432 + c], o2);
    path_tp<2,1,1>(cg + 375, x2, x1, 0.5f*mix[464 + c], o1);
    path_tp<2,1,2>(cg + 420, x2, x1, 0.5f*mix[496 + c], o2);
    path_tp<2,1,3>(cg + 495, x2, x1, 0.5f*mix[528 + c], o3);
    path_tp<2,2,0>(cg + 600, x2, x2, 0.5f*mix[560 + c], o0);
    path_tp<2,2,1>(cg + 625, x2, x2, 0.5f*mix[592 + c], o1);
    path_tp<2,2,2>(cg + 700, x2, x2, 0.5f*mix[624 + c], o2);
    path_tp<2,2,3>(cg + 825, x2, x2, 0.5f*mix[656 + c], o3);
    const float kc = keep[112 + c];                        // keep, l=2
    #pragma unroll
    for (int m = 0; m < 5; ++m) o2[m] = fmaf(kc, x2[m], o2[m]);
  }

  // Every output element is written exactly once across the 64 channel-lanes.
  float* outr = out + (size_t)row * 672;
  if (active) outr[c] = o0[0];                             // l3=0: 64 ch
  if (has1) {
    #pragma unroll
    for (int m = 0; m < 3; ++m) outr[64 + c*3 + m] = o1[m];      // l3=1: 48 ch
  }
  if (active && c < 48) {
    #pragma unroll
    for (int m = 0; m < 5; ++m) outr[208 + c*5 + m] = o2[m];     // l3=2: 48 ch
  }
  if (has2) {
    #pragma unroll
    for (int m = 0; m < 7; ++m) outr[448 + c*7 + m] = o3[m];     // l3=3: 32 ch
  }
}

// ================= launcher ====================================================
extern "C" void kernel_launch(void* const* d_in, const int* in_sizes, int n_in,
                              void* d_out, int out_size, void* d_ws, size_t ws_size,
                              hipStream_t stream) {
  (void)n_in; (void)out_size; (void)ws_size;
  const float* x    = (const float*)d_in[0];   // [32768*368]
  const float* keep = (const float*)d_in[1];   // [144]
  const float* mix  = (const float*)d_in[2];   // [688]
  float* outp = (float*)d_out;
  float* cg   = (float*)d_ws;                  // 1000 floats of scratch

  const int nrows = in_sizes[0] / 368;

  // 1) rebuild CG tables every call (deterministic; d_ws may be poisoned)
  selfmix_cg_build<<<NPATH, 64, 0, stream>>>(cg);
  // 2) main fused TP + keep kernel: 4 rows per 256-thread block (8 wave32)
  const int blocks = (nrows + 3) / 4;
  selfmix_main<<<blocks, 256, 0, stream>>>(x, keep, mix, cg, outp, nrows);
}